// GLPRModule_84799834292409
// MI455X (gfx1250) — compile-verified
//
#include <hip/hip_runtime.h>
#include <stdint.h>

// ---------------------------------------------------------------------------
// Problem constants (from the reference)
// ---------------------------------------------------------------------------
#define B_SAMPLES 256
#define FEAT_DIM  2048
#define NUM_IDS   20000
#define PROTO_FLOATS ((size_t)2 * NUM_IDS * FEAT_DIM)   // 81,920,000 per tensor
#define MOM  0.9f
#define OMM  0.1f   // 1 - momentum (f32-rounded, matches jax f32 promotion)

// Copy geometry: 20,480,000 float4 = 10,000 blocks x 256 threads x 8 float4
#define CP_BLOCKS  10000
#define CP_THREADS 256
#define CP_UNROLL  8

typedef float f4 __attribute__((ext_vector_type(4)));

// ---------------------------------------------------------------------------
// Kernel 1: streaming non-temporal B128 copy (proto_in -> proto_out region).
// Exactly one pass, no loop, no tail. 8 independent B128 NT loads issued
// back-to-back per thread (4KB in flight per wave) before the 8 NT stores,
// so latency is hidden by MLP within each wave, not just wave count.
// ---------------------------------------------------------------------------
__global__ void __launch_bounds__(CP_THREADS)
glpr_copy_nt(const f4* __restrict__ src, f4* __restrict__ dst) {
    const long base = (long)blockIdx.x * (CP_THREADS * CP_UNROLL) + threadIdx.x;

    f4 v[CP_UNROLL];
    #pragma unroll
    for (int k = 0; k < CP_UNROLL; ++k) {
        v[k] = __builtin_nontemporal_load(src + base + (long)k * CP_THREADS);
    }
    #pragma unroll
    for (int k = 0; k < CP_UNROLL; ++k) {
        __builtin_nontemporal_store(v[k], dst + base + (long)k * CP_THREADS);
    }
}

// ---------------------------------------------------------------------------
// Kernel 2: per-sample EMA update via closed-form chain.
// One block per sample i (256 threads, 8 columns each).
// Keys staged into LDS with gfx1250 async global->LDS loads.
// refined[i] = 0.7*f_i + 0.3*P_seen, where P_seen replays earlier same-key
// samples on top of the *input* prototype. Only the last occurrence of a key
// writes the final prototype row (overwriting the verbatim copy from K1).
// ---------------------------------------------------------------------------
__global__ void __launch_bounds__(B_SAMPLES)
glpr_update(const float* __restrict__ feat,
            const int*   __restrict__ modality,
            const int*   __restrict__ pids,
            const float* __restrict__ g_in,
            const float* __restrict__ l_in,
            float* __restrict__ refined,
            float* __restrict__ g_out,
            float* __restrict__ l_out) {
    __shared__ int s_mod[B_SAMPLES];
    __shared__ int s_pid[B_SAMPLES];
    __shared__ int s_key[B_SAMPLES];

    const int tid = threadIdx.x;
    const int i   = blockIdx.x;          // sample index this block owns

    // --- Async stage modality[] and pids[] into LDS (gfx1250 async path) ---
    {
        // addrspace(3) generic pointers: low 32 bits are the LDS byte offset.
        unsigned lds_mod = (unsigned)(uintptr_t)(&s_mod[tid]);
        unsigned lds_pid = (unsigned)(uintptr_t)(&s_pid[tid]);
        const int* gmod = modality + tid;
        const int* gpid = pids + tid;
        asm volatile("global_load_async_to_lds_b32 %0, %1, off"
                     :: "v"(lds_mod), "v"(gmod) : "memory");
        asm volatile("global_load_async_to_lds_b32 %0, %1, off"
                     :: "v"(lds_pid), "v"(gpid) : "memory");
        asm volatile("s_wait_asynccnt 0" ::: "memory");
    }
    __syncthreads();                     // every wave waited its own ASYNCcnt

    s_key[tid] = s_mod[tid] * NUM_IDS + s_pid[tid];
    __syncthreads();

    const int key = s_key[i];

    // Is this block's sample the last occurrence of its key?
    bool last = true;
    for (int j = i + 1; j < B_SAMPLES; ++j) {
        if (s_key[j] == key) { last = false; break; }
    }

    const float* gp = g_in + (size_t)key * FEAT_DIM;
    const float* lp = l_in + (size_t)key * FEAT_DIM;
    const float* fi_row = feat + (size_t)i * FEAT_DIM;

    #pragma unroll
    for (int k = 0; k < FEAT_DIM / B_SAMPLES; ++k) {
        const int col = tid + k * B_SAMPLES;

        float gP = gp[col];
        float lP = lp[col];

        // Replay earlier same-key samples (rare: pids are 256 draws of 20000)
        for (int j = 0; j < i; ++j) {
            if (s_key[j] == key) {
                float fj = feat[(size_t)j * FEAT_DIM + col];
                gP = MOM * gP + OMM * fj;
                lP = MOM * lP + OMM * fj;   // local_feat == feat, flat-identical
            }
        }

        float fv = fi_row[col];
        refined[(size_t)i * FEAT_DIM + col] = 0.7f * fv + 0.3f * gP;

        if (last) {
            g_out[(size_t)key * FEAT_DIM + col] = MOM * gP + OMM * fv;
            l_out[(size_t)key * FEAT_DIM + col] = MOM * lP + OMM * fv;
        }
    }
}

// ---------------------------------------------------------------------------
// Launch: copies first (bandwidth floor ~56us @ 23.3 TB/s), then the tiny
// update kernel overwrites the <=512 touched rows and writes refined.
// ---------------------------------------------------------------------------
extern "C" void kernel_launch(void* const* d_in, const int* in_sizes, int n_in,
                              void* d_out, int out_size, void* d_ws, size_t ws_size,
                              hipStream_t stream) {
    const float* feat     = (const float*)d_in[0];
    const int*   modality = (const int*)  d_in[1];
    const int*   pids     = (const int*)  d_in[2];
    const float* g_in     = (const float*)d_in[3];
    const float* l_in     = (const float*)d_in[4];

    float* refined = (float*)d_out;
    float* g_out   = refined + (size_t)B_SAMPLES * FEAT_DIM;
    float* l_out   = g_out + PROTO_FLOATS;

    // 20,480,000 float4 per tensor == CP_BLOCKS * CP_THREADS * CP_UNROLL (exact)
    glpr_copy_nt<<<CP_BLOCKS, CP_THREADS, 0, stream>>>((const f4*)g_in, (f4*)g_out);
    glpr_copy_nt<<<CP_BLOCKS, CP_THREADS, 0, stream>>>((const f4*)l_in, (f4*)l_out);

    glpr_update<<<B_SAMPLES, B_SAMPLES, 0, stream>>>(
        feat, modality, pids, g_in, l_in, refined, g_out, l_out);
}